// RNNClassifier_64759516889723
// MI455X (gfx1250) — compile-verified
//
#include <hip/hip_runtime.h>
#include <cmath>

typedef float v2f __attribute__((ext_vector_type(2)));
typedef float v8f __attribute__((ext_vector_type(8)));
typedef unsigned int u32x4 __attribute__((ext_vector_type(4)));
typedef unsigned int u32x8 __attribute__((ext_vector_type(8)));

#define B_ 256
#define T_ 256
#define I_ 2048
#define H_ 128

// -------------------------------------------------------------------------
// Tensor Data Mover: async-load a 2D tile (tileD0 x tileD1 f32 elements,
// row stride strideD0 elements in global) into LDS at ldsAddr.
// pad=true inserts 2 dwords of LDS padding after every 32 dwords
// (pad_interval code 4 = 16 x 8B = 32 dwords, pad_amount code 1 = 2 dwords),
// i.e. LDS row stride 34 floats for a 32-float tile row.
// D# layout per CDNA5 ISA. Issue from ONE wave.
// -------------------------------------------------------------------------
__device__ __forceinline__ void tdm_load_2d(unsigned ldsAddr, const void* g,
                                            unsigned tileD0, unsigned tileD1,
                                            unsigned tensorD0,
                                            unsigned long long strideD0,
                                            bool pad) {
  const unsigned long long ga = (unsigned long long)(uintptr_t)g;
  u32x4 g0;
  g0[0] = 1u;                                   // count=1 (valid), user mode
  g0[1] = ldsAddr;                              // lds_addr [63:32]
  g0[2] = (unsigned)ga;                         // global_addr lo
  g0[3] = (unsigned)(ga >> 32) | (2u << 30);    // global_addr hi | type=2
  u32x8 g1;
  g1[0] = (2u << 16)                            // data_size = 4B
        | (pad ? ((1u << 20) | (4u << 22) | (1u << 25)) : 0u);
  g1[1] = (tensorD0 & 0xFFFFu) << 16;           // tensor_dim0 lo16 (atomic addr 0)
  g1[2] = (tensorD0 >> 16);                     // tensor_dim0 hi16 | tensor_dim1 lo16(=0)
  g1[3] = (1u << 4) | (tileD0 << 16);           // tensor_dim1 = 1<<20 (hi16=16) | tile_dim0
  g1[4] = tileD1;                               // tile_dim1 | tile_dim2=0
  g1[5] = (unsigned)strideD0;                   // tensor_dim0_stride lo32
  g1[6] = (unsigned)(strideD0 >> 32) & 0xFFFFu; // stride hi16 | dim1_stride lo16=0
  g1[7] = 0u;
  asm volatile("tensor_load_to_lds %0, %1" : : "s"(g0), "s"(g1) : "memory");
}

// -------------------------------------------------------------------------
// Kernel 1: xp[b*T+t, h] = sum_i x[b,t,i] * W_ih[h,i] + (b_ih[h] + b_hh[h])
// Block = 256 threads (8 wave32) computing a 64-row x 128-col tile.
// Wave w -> row tile (w>>1)*16, column half (w&1)*64: 4 x v8f accumulators.
// K consumed in 32-wide chunks, TDM double-buffered through LDS.
// W_ih chunk stored UN-transposed: B frag (k,k+1) pair is contiguous ->
// both A and B fragments are single aligned ds_load_b64.
// -------------------------------------------------------------------------
__launch_bounds__(256)
__global__ void proj_kernel(const float* __restrict__ x,
                            const float* __restrict__ W_ih,
                            const float* __restrict__ b_ih,
                            const float* __restrict__ b_hh,
                            float* __restrict__ xp) {
  __shared__ __align__(16) float As[2][64][34];    // x rows:    64 x 32 (+2 pad)
  __shared__ __align__(16) float Ws[2][128][34];   // W_ih rows: 128 x 32 (+2 pad)

  const int tid  = threadIdx.x;
  const int lane = tid & 31;
  const int wv   = tid >> 5;
  const int mBlk = blockIdx.x * 64;

  const int rowT = (wv >> 1) << 4;     // wave's 16-row tile within block
  const int colH = (wv & 1) << 6;      // wave's 64-col half
  const int ar = lane & 15;            // A-frag row
  const int ak = (lane >> 4) << 1;     // frag K offset {0,2}
  const int cn = lane & 15;            // C/D col
  const int cm = (lane >> 4) << 3;     // C/D row base {0,8}

  v8f acc[4];
#pragma unroll
  for (int n = 0; n < 4; ++n) acc[n] = v8f{};

  if (wv == 0) {
    tdm_load_2d((unsigned)(uintptr_t)&As[0][0][0], &x[(size_t)mBlk * I_],
                32, 64, I_, I_, true);
    tdm_load_2d((unsigned)(uintptr_t)&Ws[0][0][0], W_ih,
                32, 128, I_, I_, true);
  }

  for (int c = 0; c < 64; ++c) {
    const int buf = c & 1;
    if (wv == 0) {
      if (c + 1 < 64) {                // prefetch next chunk, wait for current
        const int K1 = (c + 1) * 32;
        tdm_load_2d((unsigned)(uintptr_t)&As[buf ^ 1][0][0],
                    &x[(size_t)mBlk * I_ + K1], 32, 64, I_, I_, true);
        tdm_load_2d((unsigned)(uintptr_t)&Ws[buf ^ 1][0][0],
                    &W_ih[K1], 32, 128, I_, I_, true);
        __builtin_amdgcn_s_wait_tensorcnt(2);
      } else {
        __builtin_amdgcn_s_wait_tensorcnt(0);
      }
    }
    __syncthreads();                   // tile ready for everyone

    const int mRow = rowT + ar;
#pragma unroll
    for (int ks = 0; ks < 8; ++ks) {
      const v2f a = *(const v2f*)&As[buf][mRow][ks * 4 + ak];
#pragma unroll
      for (int n = 0; n < 4; ++n) {
        const v2f b = *(const v2f*)&Ws[buf][colH + n * 16 + cn][ks * 4 + ak];
        acc[n] = __builtin_amdgcn_wmma_f32_16x16x4_f32(
            false, a, false, b, (short)0, acc[n], false, false);
      }
    }
    __syncthreads();                   // done reading before buffer reuse
  }

  const int wRow0 = mBlk + rowT;
#pragma unroll
  for (int n = 0; n < 4; ++n) {
    const int col  = colH + n * 16 + cn;
    const float bs = b_ih[col] + b_hh[col];
#pragma unroll
    for (int v = 0; v < 8; ++v) {
      xp[(size_t)(wRow0 + cm + v) * H_ + col] = acc[n][v] + bs;
    }
  }
}

// -------------------------------------------------------------------------
// Kernel 2: h = tanh(xp_t + h @ W_hh^T), 256 sequential steps.
// Batch rows independent -> 16 blocks x 16 rows, no inter-block sync.
// Block = 256 threads (8 waves); wave w owns column tile [16w, 16w+16).
// W_hh^T B-fragments are loop-invariant: preloaded into 64 VGPRs once.
// The per-step 16x128 xp tile is TDM-prefetched one step ahead, so the
// sequential hot loop has zero per-lane global loads.
// -------------------------------------------------------------------------
__launch_bounds__(256)
__global__ void rnn_kernel(const float* __restrict__ xp,
                           const float* __restrict__ W_hh,
                           float* __restrict__ hT) {
  __shared__ __align__(16) float hs[16][132];     // h slab (8B-aligned pairs)
  __shared__ __align__(16) float xs[2][16][128];  // double-buffered xp tile

  const int tid  = threadIdx.x;
  const int lane = tid & 31;
  const int wv   = tid >> 5;
  const int b0   = blockIdx.x * 16;

  const int ar = lane & 15;
  const int ak = (lane >> 4) << 1;
  const int cn = lane & 15;
  const int cm = (lane >> 4) << 3;
  const int n0 = wv << 4;

  // Preload this wave's 32 B-fragments: B(k, N) = W_hh[N][k], k = 4*ks + ak.
  v2f bf[32];
#pragma unroll
  for (int ks = 0; ks < 32; ++ks) {
    bf[ks] = *(const v2f*)&W_hh[(size_t)(n0 + cn) * H_ + ks * 4 + ak];
  }

  for (int j = tid; j < 16 * 132; j += 256) hs[j / 132][j % 132] = 0.0f;

  if (wv == 0) {                       // prefetch t=0 xp tile (rows stride T*H)
    tdm_load_2d((unsigned)(uintptr_t)&xs[0][0][0], &xp[(size_t)b0 * T_ * H_],
                H_, 16, H_, (unsigned long long)T_ * H_, false);
  }
  __syncthreads();

  for (int t = 0; t < T_; ++t) {
    if (wv == 0) {
      if (t + 1 < T_) {                // prefetch next step, wait for current
        tdm_load_2d((unsigned)(uintptr_t)&xs[(t + 1) & 1][0][0],
                    &xp[((size_t)b0 * T_ + (t + 1)) * H_],
                    H_, 16, H_, (unsigned long long)T_ * H_, false);
        __builtin_amdgcn_s_wait_tensorcnt(1);
      } else {
        __builtin_amdgcn_s_wait_tensorcnt(0);
      }
    }
    __syncthreads();                   // xs[t&1] ready; hs writes visible

    v8f acc0 = v8f{};                  // two chains for XDL ILP
    v8f acc1 = v8f{};
#pragma unroll
    for (int ks = 0; ks < 32; ks += 2) {
      const v2f a0 = *(const v2f*)&hs[ar][ks * 4 + ak];
      const v2f a1 = *(const v2f*)&hs[ar][ks * 4 + 4 + ak];
      acc0 = __builtin_amdgcn_wmma_f32_16x16x4_f32(
          false, a0, false, bf[ks], (short)0, acc0, false, false);
      acc1 = __builtin_amdgcn_wmma_f32_16x16x4_f32(
          false, a1, false, bf[ks + 1], (short)0, acc1, false, false);
    }
    float hnew[8];
#pragma unroll
    for (int v = 0; v < 8; ++v) {
      hnew[v] = tanhf(acc0[v] + acc1[v] + xs[t & 1][cm + v][n0 + cn]);
    }
    __syncthreads();                   // all reads of hs / xs[t&1] done
#pragma unroll
    for (int v = 0; v < 8; ++v) hs[cm + v][n0 + cn] = hnew[v];
  }
  __syncthreads();

  for (int j = tid; j < 16 * H_; j += 256)
    hT[(size_t)(b0 + (j >> 7)) * H_ + (j & 127)] = hs[j >> 7][j & 127];
}

// -------------------------------------------------------------------------
// Kernel 3: out[b] = hT[b,:] . W_fc + b_fc   (tiny)
// -------------------------------------------------------------------------
__global__ void fc_kernel(const float* __restrict__ hT,
                          const float* __restrict__ W_fc,
                          const float* __restrict__ b_fc,
                          float* __restrict__ out) {
  const int b = blockIdx.x * blockDim.x + threadIdx.x;
  if (b < B_) {
    float s = b_fc[0];
#pragma unroll 8
    for (int k = 0; k < H_; ++k) s += hT[(size_t)b * H_ + k] * W_fc[k];
    out[b] = s;
  }
}

extern "C" void kernel_launch(void* const* d_in, const int* in_sizes, int n_in,
                              void* d_out, int out_size, void* d_ws, size_t ws_size,
                              hipStream_t stream) {
  (void)in_sizes; (void)n_in; (void)out_size; (void)ws_size;
  const float* x    = (const float*)d_in[0];
  const float* W_ih = (const float*)d_in[1];
  const float* W_hh = (const float*)d_in[2];
  const float* b_ih = (const float*)d_in[3];
  const float* b_hh = (const float*)d_in[4];
  const float* W_fc = (const float*)d_in[5];
  const float* b_fc = (const float*)d_in[6];
  float* out = (float*)d_out;

  float* xp = (float*)d_ws;                          // B*T*H f32 = 32 MiB
  float* hT = xp + (size_t)B_ * T_ * H_;             // B*H f32 = 128 KiB

  proj_kernel<<<(B_ * T_) / 64, 256, 0, stream>>>(x, W_ih, b_ih, b_hh, xp);
  rnn_kernel<<<B_ / 16, 256, 0, stream>>>(xp, W_hh, hT);
  fc_kernel<<<1, 256, 0, stream>>>(hT, W_fc, b_fc, out);
}